// FISLoss_25460566131021
// MI455X (gfx1250) — compile-verified
//
#include <hip/hip_runtime.h>
#include <math.h>

#define TPB1 256
#define TILE_ELEMS (TPB1 * 4)   // 1024 floats = 4KB per tile
#define NUM_GROUPS 64
#define WARMUP_EPOCHS 3
#define NEG_INF (-__builtin_inff())

// ---------------- CDNA5 async global->LDS path ----------------

typedef int v4i __attribute__((ext_vector_type(4)));
typedef __attribute__((address_space(1))) v4i gv4i_t;   // global <4 x i32>
typedef __attribute__((address_space(3))) v4i lv4i_t;   // LDS    <4 x i32>

__device__ __forceinline__ void async_copy_b128(const float* gsrc, float* ldst) {
#if __has_builtin(__builtin_amdgcn_global_load_async_to_lds_b128)
  __builtin_amdgcn_global_load_async_to_lds_b128(
      (gv4i_t*)gsrc, (lv4i_t*)ldst, /*imm_offset=*/0, /*cpol=*/0);
#else
  unsigned loff =
      (unsigned)(unsigned long long)(__attribute__((address_space(3))) char*)ldst;
  asm volatile("global_load_async_to_lds_b128 %0, %1, off"
               :: "v"(loff), "v"(gsrc)
               : "memory");
#endif
}

#if __has_builtin(__builtin_amdgcn_s_wait_asynccnt)
#define WAIT_ASYNC(n) __builtin_amdgcn_s_wait_asynccnt(n)
#else
#define WAIT_ASYNC(n) asm volatile("s_wait_asynccnt %0" :: "i"(n) : "memory")
#endif

#define WAIT_DS0() asm volatile("s_wait_dscnt 0x0" ::: "memory")

// ---------------- Kernel 1: per-row online logsumexp -> losses[N] ----------------

// Grouped online-logsumexp update: one (rare) rescale branch per 4 elements,
// 4 independent v_exp_f32 on the common path.
__device__ __forceinline__ void lse_update4(float4 v, float& m, float& s) {
  float vmax = fmaxf(fmaxf(v.x, v.y), fmaxf(v.z, v.w));
  if (vmax > m) {                 // rare for N(0,1) data after warm-up
    s *= __expf(m - vmax);
    m = vmax;
  }
  s += __expf(v.x - m) + __expf(v.y - m) + __expf(v.z - m) + __expf(v.w - m);
}

__device__ __forceinline__ void lse_update1(float x, float& m, float& s) {
  if (x > m) {
    s = s * __expf(m - x) + 1.0f;
    m = x;
  } else {
    s += __expf(x - m);
  }
}

__global__ __launch_bounds__(TPB1) void ce_row_kernel(
    const float* __restrict__ logits, const int* __restrict__ targets,
    float* __restrict__ losses, int C) {
  __shared__ float buf[2][TILE_ELEMS];                 // 8 KB double buffer
  __shared__ float red_m[TPB1 / 32], red_s[TPB1 / 32]; // cross-wave combine

  const int row = blockIdx.x;
  const int tid = threadIdx.x;
  const float* rp = logits + (size_t)row * (size_t)C;

  const int NT = C / TILE_ELEMS;   // full tiles
  float m = NEG_INF, s = 0.0f;

  if (NT > 0) {
    async_copy_b128(rp + tid * 4, &buf[0][tid * 4]);

    // steady state: issue t+1, wait for t, consume t  (no branch in body)
    for (int t = 0; t < NT - 1; ++t) {
      WAIT_DS0();   // close WAR window on buf[(t+1)&1] (read in the previous iter)
      async_copy_b128(rp + (size_t)(t + 1) * TILE_ELEMS + tid * 4,
                      &buf[(t + 1) & 1][tid * 4]);
      WAIT_ASYNC(1);   // async loads complete in order -> tile t is in LDS
      lse_update4(((const float4*)buf[t & 1])[tid], m, s);
    }
    // epilogue: last tile
    WAIT_ASYNC(0);
    lse_update4(((const float4*)buf[(NT - 1) & 1])[tid], m, s);
  }
  // scalar tail (C % 1024 elements)
  for (int j = NT * TILE_ELEMS + tid; j < C; j += TPB1) {
    lse_update1(rp[j], m, s);
  }

  // wave32 reduce of (m, s)
  #pragma unroll
  for (int o = 16; o; o >>= 1) {
    float mo = __shfl_xor(m, o, 32);
    float so = __shfl_xor(s, o, 32);
    float mn = fmaxf(m, mo);
    s = s * __expf(m - mn) + so * __expf(mo - mn);
    m = mn;
  }
  const int lane = tid & 31, wid = tid >> 5;
  if (lane == 0) { red_m[wid] = m; red_s[wid] = s; }
  __syncthreads();
  if (tid == 0) {
    float M = red_m[0], S = red_s[0];
    #pragma unroll
    for (int w = 1; w < TPB1 / 32; ++w) {
      float mw = red_m[w], sw = red_s[w];
      float mn = fmaxf(M, mw);
      S = S * __expf(M - mn) + sw * __expf(mw - mn);
      M = mn;
    }
    const int tg = targets[row];
    losses[row] = (M + __logf(S)) - rp[tg];
  }
}

// ---------------- Kernel 2: group stats + reweighted mean -> out[0] ----------------

__global__ __launch_bounds__(1024) void finalize_kernel(
    const float* __restrict__ losses, const int* __restrict__ groups,
    const int* __restrict__ epoch_p, float* __restrict__ out, int N) {
  __shared__ float gsum[NUM_GROUPS], gcnt[NUM_GROUPS], gmean[NUM_GROUPS];
  __shared__ float wred[32], wred2[32];
  __shared__ float sh_mean, sh_maxl;

  const int tid = threadIdx.x;
  const int nthreads = blockDim.x;
  if (tid < NUM_GROUPS) { gsum[tid] = 0.0f; gcnt[tid] = 0.0f; }
  __syncthreads();

  // pass A: total + per-group sums
  float tot = 0.0f;
  for (int i = tid; i < N; i += nthreads) {
    float l = losses[i];
    int g = groups[i];
    atomicAdd(&gsum[g], l);
    atomicAdd(&gcnt[g], 1.0f);
    tot += l;
  }
  #pragma unroll
  for (int o = 16; o; o >>= 1) tot += __shfl_xor(tot, o, 32);
  if ((tid & 31) == 0) wred[tid >> 5] = tot;
  __syncthreads();
  if (tid < 32) {
    float v = wred[tid];
    #pragma unroll
    for (int o = 16; o; o >>= 1) v += __shfl_xor(v, o, 32);
    if (tid == 0) sh_mean = v / (float)N;
  }
  __syncthreads();

  if (tid < NUM_GROUPS) gmean[tid] = gsum[tid] / fmaxf(gcnt[tid], 1.0f);
  __syncthreads();
  if (tid == 0) {
    float mx = NEG_INF;
    for (int g = 0; g < NUM_GROUPS; ++g)
      if (gcnt[g] > 0.0f) mx = fmaxf(mx, gmean[g]);
    sh_maxl = mx + 1e-6f;
  }
  __syncthreads();

  const int epoch = epoch_p[0];
  const float mean = sh_mean;
  if (epoch < WARMUP_EPOCHS) {
    if (tid == 0) out[0] = mean;
    return;
  }
  const float progress = fminf(1.0f, (float)(epoch - WARMUP_EPOCHS) / 30.0f);
  const float alpha = 0.8f + progress * (0.2f - 0.8f);
  const float maxl = sh_maxl;

  // pass B: weighted sums
  float s1 = 0.0f, s2 = 0.0f;
  for (int i = tid; i < N; i += nthreads) {
    float l = losses[i];
    int g = groups[i];
    float ind = 1.0f / (1.0f + __expf(mean - l));        // sigmoid(l - mean)
    ind = fminf(fmaxf(ind, 0.1f), 10.0f);
    float gw = gmean[g] / maxl;
    float fw = alpha * ind + (1.0f - alpha) * gw;
    fw = fminf(fmaxf(fw, 0.01f), 10.0f);
    s1 += fw * l;
    s2 += fw;
  }
  #pragma unroll
  for (int o = 16; o; o >>= 1) {
    s1 += __shfl_xor(s1, o, 32);
    s2 += __shfl_xor(s2, o, 32);
  }
  if ((tid & 31) == 0) { wred[tid >> 5] = s1; wred2[tid >> 5] = s2; }
  __syncthreads();
  if (tid < 32) {
    float a = wred[tid], b = wred2[tid];
    #pragma unroll
    for (int o = 16; o; o >>= 1) {
      a += __shfl_xor(a, o, 32);
      b += __shfl_xor(b, o, 32);
    }
    if (tid == 0) out[0] = fminf(a / b, 10.0f);   // mean(fw*l)/mean(fw) = Σ/Σ
  }
}

// ---------------- launch ----------------

extern "C" void kernel_launch(void* const* d_in, const int* in_sizes, int n_in,
                              void* d_out, int out_size, void* d_ws, size_t ws_size,
                              hipStream_t stream) {
  const float* logits = (const float*)d_in[0];
  const int* targets  = (const int*)d_in[1];
  const int* groups   = (const int*)d_in[2];
  const int* epoch    = (const int*)d_in[3];
  float* out = (float*)d_out;
  float* losses = (float*)d_ws;   // N floats of scratch

  const int N = in_sizes[1];
  const int C = in_sizes[0] / N;

  hipLaunchKernelGGL(ce_row_kernel, dim3(N), dim3(TPB1), 0, stream,
                     logits, targets, losses, C);
  hipLaunchKernelGGL(finalize_kernel, dim3(1), dim3(1024), 0, stream,
                     losses, groups, epoch, out, N);
}